// mLSTMCell_23003844838052
// MI455X (gfx1250) — compile-verified
//
#include <hip/hip_runtime.h>
#include <math.h>

#define Bsz  64
#define Hdim 1024

typedef float v2f __attribute__((ext_vector_type(2)));
typedef float v8f __attribute__((ext_vector_type(8)));

// ---------------- K1: 1D conv (k=4, SAME: pad_left=1) + SiLU -> qk ----------
__global__ void conv_silu_kernel(const float* __restrict__ x,
                                 const float* __restrict__ cw,
                                 const float* __restrict__ cb,
                                 float* __restrict__ qk) {
    int idx = blockIdx.x * blockDim.x + threadIdx.x;   // over B*H
    int b = idx >> 10;
    int w = idx & 1023;
    const float* xr = x + (size_t)b * Hdim;
    float s = cb[0];
#pragma unroll
    for (int k = 0; k < 4; ++k) {
        int j = w + k - 1;
        float xv = (j >= 0 && j < Hdim) ? xr[j] : 0.0f;
        s = fmaf(xv, cw[k], s);
    }
    qk[idx] = s / (1.0f + __expf(-s));                 // silu
}

// ---------------- K2: i/f gate matvecs + m_new, i_g, f_g --------------------
__global__ void gates_kernel(const float* __restrict__ x,
                             const float* __restrict__ Wi, const float* __restrict__ bi,
                             const float* __restrict__ Wf, const float* __restrict__ bf,
                             const float* __restrict__ m,
                             float* __restrict__ m_new,
                             float* __restrict__ ig, float* __restrict__ fg) {
    __shared__ float si[256], sf[256];
    int b = blockIdx.x;
    int t = threadIdx.x;
    const float* xr = x + (size_t)b * Hdim;
    float ai = 0.f, af = 0.f;
    for (int k = t; k < Hdim; k += 256) {
        float xv = xr[k];
        ai = fmaf(xv, Wi[k], ai);
        af = fmaf(xv, Wf[k], af);
    }
    si[t] = ai; sf[t] = af;
    __syncthreads();
    for (int off = 128; off > 0; off >>= 1) {
        if (t < off) { si[t] += si[t + off]; sf[t] += sf[t + off]; }
        __syncthreads();
    }
    if (t == 0) {
        float it = si[0] + bi[0];
        float ft = sf[0] + bf[0];
        float mo = m[b];
        float mn = fmaxf(ft + mo, it);
        m_new[b] = mn;
        ig[b] = __expf(it - mn);
        fg[b] = __expf(ft + mo - mn);
    }
}

// ---------------- K3: fp32 WMMA GEMM 64x1024 @ 1024x1024 + bias + epilogue --
// EPI: 0 = none, 1 = sigmoid, 2 = * 1/sqrt(H) = 1/32
template <int EPI>
__global__ void __launch_bounds__(256) wmma_gemm_64x1024(
        const float* __restrict__ A,     // 64 x 1024 row-major
        const float* __restrict__ W,     // 1024 x 1024 row-major (K x N)
        const float* __restrict__ bias,  // 1024
        float* __restrict__ Y) {         // 64 x 1024
    const int N = Hdim, K = Hdim;
    int wave = (blockIdx.x * blockDim.x + threadIdx.x) >> 5;  // 0..255
    int lane = threadIdx.x & 31;
    int tile_m = wave & 3;      // 4 row tiles (M=64)
    int tile_n = wave >> 2;     // 64 col tiles (N=1024)
    int half = lane >> 4;       // 0 or 1
    int lm   = lane & 15;

    const float* arow = A + (size_t)(tile_m * 16 + lm) * K;
    const float* wcol = W + tile_n * 16 + lm;

    v8f acc = {0.f, 0.f, 0.f, 0.f, 0.f, 0.f, 0.f, 0.f};
    for (int k0 = 0; k0 < K; k0 += 4) {
        int ka = k0 + 2 * half;                 // this lane-half's K pair
        v2f a = *(const v2f*)(arow + ka);       // A[m, ka], A[m, ka+1]
        v2f bfrag;
        bfrag.x = wcol[(size_t)ka * N];         // W[ka,   n]
        bfrag.y = wcol[(size_t)(ka + 1) * N];   // W[ka+1, n]
        acc = __builtin_amdgcn_wmma_f32_16x16x4_f32(
            /*neg_a=*/false, a, /*neg_b=*/false, bfrag,
            /*c_mod=*/(short)0, acc, /*reuse_a=*/false, /*reuse_b=*/false);
    }

    int n = tile_n * 16 + lm;
    float bb = bias[n];
#pragma unroll
    for (int r = 0; r < 8; ++r) {
        int mrow = tile_m * 16 + r + 8 * half;  // C/D layout: M = vgpr + 8*half
        float v = acc[r] + bb;
        if (EPI == 1)      v = 1.0f / (1.0f + __expf(-v));
        else if (EPI == 2) v *= 0.03125f;       // 1/sqrt(1024)
        Y[(size_t)mrow * N + n] = v;
    }
}

// ---------------- K4: n_new = f_g*n + i_g*key ; scaler = 1/max(|n_new.q|,1) -
__global__ void norm_scaler_kernel(const float* __restrict__ n_old,
                                   const float* __restrict__ keyv,
                                   const float* __restrict__ qry,
                                   const float* __restrict__ ig,
                                   const float* __restrict__ fg,
                                   float* __restrict__ n_new,
                                   float* __restrict__ scaler) {
    __shared__ float sred[256];
    int b = blockIdx.x;
    int t = threadIdx.x;
    float fgb = fg[b], igb = ig[b];
    float part = 0.f;
    for (int k = t; k < Hdim; k += 256) {
        size_t idx = (size_t)b * Hdim + k;
        float nn = fmaf(fgb, n_old[idx], igb * keyv[idx]);
        n_new[idx] = nn;
        part = fmaf(nn, qry[idx], part);
    }
    sred[t] = part;
    __syncthreads();
    for (int off = 128; off > 0; off >>= 1) {
        if (t < off) sred[t] += sred[t + off];
        __syncthreads();
    }
    if (t == 0) {
        float s = fabsf(sred[0]);
        scaler[b] = 1.0f / fmaxf(s, 1.0f);
    }
}

// ---------------- K5: fused c_new stream + matvec with query + h ------------
// One wave per (b,i) row of c. 8 waves/block share one batch's q/key in LDS.
__global__ void __launch_bounds__(256) cell_update_kernel(
        const float* __restrict__ c_old,
        const float* __restrict__ keyv,
        const float* __restrict__ qry,
        const float* __restrict__ val,
        const float* __restrict__ outg,
        const float* __restrict__ skip,
        const float* __restrict__ ig,
        const float* __restrict__ fg,
        const float* __restrict__ scaler,
        float* __restrict__ c_new,
        float* __restrict__ h) {
    __shared__ float q_s[Hdim];
    __shared__ float k_s[Hdim];
    int waveId = threadIdx.x >> 5;
    int lane   = threadIdx.x & 31;
    int rowBase = blockIdx.x * 8;        // 8 rows per block; 8 | 1024 -> one batch per block
    int b = rowBase >> 10;

    for (int k = threadIdx.x; k < Hdim; k += 256) {
        q_s[k] = qry[(size_t)b * Hdim + k];
        k_s[k] = keyv[(size_t)b * Hdim + k];
    }
    __syncthreads();

    int row = rowBase + waveId;          // global row in [0, B*H)
    float fgb = fg[b];
    float igv = ig[b] * val[row];

    const float4* csrc = (const float4*)(c_old + (size_t)row * Hdim);
    float4*       cdst = (float4*)(c_new + (size_t)row * Hdim);
    const float4* k4p  = (const float4*)k_s;
    const float4* q4p  = (const float4*)q_s;

    float acc = 0.f;
#pragma unroll
    for (int ch = 0; ch < 8; ++ch) {
        int k4 = ch * 32 + lane;         // 32 lanes x float4 = 128 floats/iter
        float4 c4   = csrc[k4];
        float4 key4 = k4p[k4];
        float4 q4   = q4p[k4];
        float4 cn;
        cn.x = fmaf(fgb, c4.x, igv * key4.x);
        cn.y = fmaf(fgb, c4.y, igv * key4.y);
        cn.z = fmaf(fgb, c4.z, igv * key4.z);
        cn.w = fmaf(fgb, c4.w, igv * key4.w);
        cdst[k4] = cn;
        acc = fmaf(cn.x, q4.x, acc);
        acc = fmaf(cn.y, q4.y, acc);
        acc = fmaf(cn.z, q4.z, acc);
        acc = fmaf(cn.w, q4.w, acc);
    }
#pragma unroll
    for (int off = 16; off > 0; off >>= 1)
        acc += __shfl_xor(acc, off, 32);
    if (lane == 0)
        h[row] = fmaf(outg[row] * acc, scaler[b], skip[row]);
}

extern "C" void kernel_launch(void* const* d_in, const int* in_sizes, int n_in,
                              void* d_out, int out_size, void* d_ws, size_t ws_size,
                              hipStream_t stream) {
    const float* c   = (const float*)d_in[0];
    const float* n   = (const float*)d_in[1];
    const float* m   = (const float*)d_in[2];
    const float* x   = (const float*)d_in[3];
    const float* Wq  = (const float*)d_in[4];
    const float* bq  = (const float*)d_in[5];
    const float* Wk  = (const float*)d_in[6];
    const float* bk  = (const float*)d_in[7];
    const float* Wv  = (const float*)d_in[8];
    const float* bv  = (const float*)d_in[9];
    const float* cw  = (const float*)d_in[10];
    const float* cb  = (const float*)d_in[11];
    const float* Wi  = (const float*)d_in[12];
    const float* bi  = (const float*)d_in[13];
    const float* Wf  = (const float*)d_in[14];
    const float* bf  = (const float*)d_in[15];
    const float* Wo  = (const float*)d_in[16];
    const float* bo  = (const float*)d_in[17];
    const float* Wsk = (const float*)d_in[18];
    const float* bsk = (const float*)d_in[19];

    // output layout: [c_new | n_new | m_new | h]
    float* c_new = (float*)d_out;
    float* n_new = c_new + (size_t)Bsz * Hdim * Hdim;
    float* m_new = n_new + (size_t)Bsz * Hdim;
    float* h     = m_new + Bsz;

    // workspace layout (floats)
    float* ws = (float*)d_ws;
    const size_t BH = (size_t)Bsz * Hdim;
    float* qk     = ws;
    float* outg   = ws + BH;
    float* val    = ws + 2 * BH;
    float* qry    = ws + 3 * BH;
    float* keyv   = ws + 4 * BH;
    float* skip   = ws + 5 * BH;
    float* ig     = ws + 6 * BH;
    float* fg     = ig + Bsz;
    float* scaler = fg + Bsz;

    conv_silu_kernel<<<(Bsz * Hdim) / 256, 256, 0, stream>>>(x, cw, cb, qk);
    gates_kernel<<<Bsz, 256, 0, stream>>>(x, Wi, bi, Wf, bf, m, m_new, ig, fg);

    const int gemm_blocks = (Bsz / 16) * (Hdim / 16) / 8;  // 256 waves, 8/block
    wmma_gemm_64x1024<1><<<gemm_blocks, 256, 0, stream>>>(x,  Wo,  bo,  outg);
    wmma_gemm_64x1024<0><<<gemm_blocks, 256, 0, stream>>>(x,  Wv,  bv,  val);
    wmma_gemm_64x1024<0><<<gemm_blocks, 256, 0, stream>>>(qk, Wq,  bq,  qry);
    wmma_gemm_64x1024<2><<<gemm_blocks, 256, 0, stream>>>(qk, Wk,  bk,  keyv);
    wmma_gemm_64x1024<0><<<gemm_blocks, 256, 0, stream>>>(qk, Wsk, bsk, skip);

    norm_scaler_kernel<<<Bsz, 256, 0, stream>>>(n, keyv, qry, ig, fg, n_new, scaler);

    cell_update_kernel<<<(Bsz * Hdim) / 8, 256, 0, stream>>>(
        c, keyv, qry, val, outg, skip, ig, fg, scaler, c_new, h);
}